// InverseDaubechiesWaveletLayer_59571196396019
// MI455X (gfx1250) — compile-verified
//
#include <hip/hip_runtime.h>

// Inverse db4 wavelet reconstruction on MI455X (gfx1250).
//
// Bandwidth-bound problem (~2 FLOP/byte, ~11us floor at 23.3 TB/s), so the
// kernel is organized around streaming: each wave computes 16x16 f32 output
// tiles via V_WMMA_F32_16X16X4_F32, expressing the 8-tap dual-filter
// polyphase convolution as a banded (16x12)x(12x16) matmul whose A-operand
// (filter band) is tile-invariant and built once per wave.
//
// Layouts (CDNA5 ISA 7.12.2, wave32):
//   A 16x4 f32 : lane m = L&15; VGPR0 K = (L<16?0:2), VGPR1 K = (L<16?1:3)
//   B 4x16 f32 : lane n = L&15; VGPR0 K = (L<16?0:2), VGPR1 K = (L<16?1:3)
//   C/D 16x16  : 8 VGPRs; VGPR v holds M = v + (L<16?0:8), N = L&15

typedef __attribute__((ext_vector_type(2))) float v2f;
typedef __attribute__((ext_vector_type(8))) float v8f;

#define NB    16          // batch
#define LIN   16000       // input time length
#define CIN   128         // input channels (64 approx + 64 detail)
#define CH    64          // output channels
#define LOUT  32000       // output time length
#define TILES_PER_WAVE 8  // consecutive 16-wide time tiles per wave
#define TIME_TILES (LOUT / 16)                  // 2000
#define STRIPS     (TIME_TILES / TILES_PER_WAVE) // 250
#define CTILES     (CH / 16)                     // 4
#define TOTAL_WAVES (NB * CTILES * STRIPS)       // 16000
#define WAVES_PER_BLOCK 8                        // 256 threads, wave32

__device__ __forceinline__ float ld_in(const float* __restrict__ in,
                                       int bb, int j, int ch) {
  // Zero-padded read of inputs[bb, j, ch]; j may fall outside [0, LIN).
  const bool ok = (j >= 0) && (j < LIN);
  const int jc = ok ? j : 0;
  const float v = in[(size_t)(bb * LIN + jc) * CIN + ch];
  return ok ? v : 0.0f;
}

__global__ __launch_bounds__(256)
void idwt_db4_wmma(const float* __restrict__ in,
                   const float* __restrict__ rec_lo,
                   const float* __restrict__ rec_hi,
                   float* __restrict__ out) {
  const int wave = blockIdx.x * WAVES_PER_BLOCK + (threadIdx.x >> 5);
  const int lane = threadIdx.x & 31;

  // ct fastest so the 8 waves of a block cover all 64 channels of 2 batches
  // for the same time strip -> contiguous 256B rows hit together in L2.
  const int ct    = wave & (CTILES - 1);
  const int bb    = (wave >> 2) & (NB - 1);
  const int strip = wave >> 6;

  const int c0 = ct * 16;
  const int m  = lane & 15;          // A: row M / B: col N / D: col N
  const bool hi = lane >= 16;
  const int kx = hi ? 2 : 0;         // K slot held in vector component .x
  const int ky = hi ? 3 : 1;         // K slot held in vector component .y

  // Tile-invariant A fragments: F[m, jj] = f[2*jj - m + 1] if tap in [0,8).
  // jj = 4*chunk + K-slot; K=12 band covers the 11 contributing input rows.
  v2f Alo[3], Ahi[3];
#pragma unroll
  for (int cc = 0; cc < 3; ++cc) {
    const int jjx = 4 * cc + kx, jjy = 4 * cc + ky;
    const int tfx = 2 * jjx - m + 1, tfy = 2 * jjy - m + 1;
    Alo[cc].x = ((unsigned)tfx < 8u) ? rec_lo[tfx] : 0.0f;
    Alo[cc].y = ((unsigned)tfy < 8u) ? rec_lo[tfy] : 0.0f;
    Ahi[cc].x = ((unsigned)tfx < 8u) ? rec_hi[tfx] : 0.0f;
    Ahi[cc].y = ((unsigned)tfy < 8u) ? rec_hi[tfy] : 0.0f;
  }

  const int chA = c0 + m;        // approx channel for this lane's B column
  const int chD = CH + c0 + m;   // detail channel

#pragma unroll 1
  for (int ti = 0; ti < TILES_PER_WAVE; ++ti) {
    const int T0    = (strip * TILES_PER_WAVE + ti) * 16;  // even
    const int jbase = (T0 >> 1) - 1;                       // first input row

    v8f acc = {};
#pragma unroll
    for (int cc = 0; cc < 3; ++cc) {
      const int jx = jbase + 4 * cc + kx;
      const int jy = jbase + 4 * cc + ky;
      v2f Ba, Bd;
      Ba.x = ld_in(in, bb, jx, chA);
      Ba.y = ld_in(in, bb, jy, chA);
      Bd.x = ld_in(in, bb, jx, chD);
      Bd.y = ld_in(in, bb, jy, chD);
      // D = A*B + C, f32 in / f32 out (8 args: neg_a,A,neg_b,B,c_mod,C,reuseA,reuseB)
      acc = __builtin_amdgcn_wmma_f32_16x16x4_f32(
          false, Alo[cc], false, Ba, (short)0, acc, false, false);
      acc = __builtin_amdgcn_wmma_f32_16x16x4_f32(
          false, Ahi[cc], false, Bd, (short)0, acc, false, false);
    }

    // Store D: VGPR v -> row T0 + v + (hi?8:0), col c0+m. Streaming (NT) stores.
    const int rbase = T0 + (hi ? 8 : 0);
    float* op = out + ((size_t)bb * LOUT + rbase) * CH + (c0 + m);
#pragma unroll
    for (int v = 0; v < 8; ++v) {
      __builtin_nontemporal_store(acc[v], op + (size_t)v * CH);
    }
  }
}

extern "C" void kernel_launch(void* const* d_in, const int* in_sizes, int n_in,
                              void* d_out, int out_size, void* d_ws, size_t ws_size,
                              hipStream_t stream) {
  (void)in_sizes; (void)n_in; (void)out_size; (void)d_ws; (void)ws_size;
  const float* in     = (const float*)d_in[0];  // (16,16000,128) f32
  const float* rec_lo = (const float*)d_in[1];  // (8,) f32
  const float* rec_hi = (const float*)d_in[2];  // (8,) f32
  float* out          = (float*)d_out;          // (16,32000,64) f32

  const int blocks = TOTAL_WAVES / WAVES_PER_BLOCK;  // 2000
  idwt_db4_wmma<<<blocks, WAVES_PER_BLOCK * 32, 0, stream>>>(in, rec_lo, rec_hi, out);
}